// TextSAGE_38912403702074
// MI455X (gfx1250) — compile-verified
//
#include <hip/hip_runtime.h>

#define DIM 64          // latent dim
#define DIM2 128        // concat(self, agg) width

typedef __attribute__((ext_vector_type(2))) float v2f;
typedef __attribute__((ext_vector_type(8))) float v8f;

// ---------------------------------------------------------------------------
// Utility: zero-fill float buffer
// ---------------------------------------------------------------------------
__global__ void fill_zero_kernel(float* __restrict__ p, long count) {
    long stride = (long)gridDim.x * blockDim.x;
    for (long i = (long)blockIdx.x * blockDim.x + threadIdx.x; i < count; i += stride)
        p[i] = 0.0f;
}

// ---------------------------------------------------------------------------
// Degree histogram: deg[dst[e]] += 1
// ---------------------------------------------------------------------------
__global__ void degree_kernel(const int* __restrict__ dst, float* __restrict__ deg,
                              int nedges) {
    int stride = gridDim.x * blockDim.x;
    for (int e = blockIdx.x * blockDim.x + threadIdx.x; e < nedges; e += stride)
        atomicAdd(&deg[dst[e]], 1.0f);
}

// inv_deg[i] = 1 / max(deg[i], 1)  (in place)
__global__ void invdeg_kernel(float* __restrict__ deg, int n) {
    int stride = gridDim.x * blockDim.x;
    for (int i = blockIdx.x * blockDim.x + threadIdx.x; i < n; i += stride)
        deg[i] = 1.0f / fmaxf(deg[i], 1.0f);
}

// ---------------------------------------------------------------------------
// Scatter-add: agg[dst[e], :] += h[src[e], :]
// 16 lanes per edge, float4 per lane (64 floats/row). Gathers are 256B per
// half-wave (coalesced); adds go through L2 atomics (working set < L2).
// ---------------------------------------------------------------------------
__global__ void scatter_kernel(const float* __restrict__ h,
                               const int* __restrict__ src,
                               const int* __restrict__ dst,
                               float* __restrict__ agg, int nedges) {
    long total  = (long)nedges * 16;            // 16 float4 chunks per edge
    long stride = (long)gridDim.x * blockDim.x;
    for (long t = (long)blockIdx.x * blockDim.x + threadIdx.x; t < total; t += stride) {
        int e = (int)(t >> 4);
        int c = (int)(t & 15);
        int s = src[e];
        int d = dst[e];
        const float4 v = *(const float4*)(h + (size_t)s * DIM + c * 4);
        float* p = agg + (size_t)d * DIM + c * 4;
        atomicAdd(p + 0, v.x);
        atomicAdd(p + 1, v.y);
        atomicAdd(p + 2, v.z);
        atomicAdd(p + 3, v.w);
    }
}

// ---------------------------------------------------------------------------
// Fused SAGE linear: out = relu?( concat(h, agg*inv_deg) @ W + b )
//   h, agg: [n, 64]   W: [128, 64]   b: [64]   out: [n, 64]
// Block = 256 threads = 8 waves. Each wave computes a 16x64 output tile via
// 32 chained V_WMMA_F32_16X16X4_F32 per accumulator column (4 accumulators).
//
// W is staged in LDS re-packed as K-pair float2s:
//   Wp[(k/2)*64 + n] = { W[k][n], W[k+1][n] }   (k even)
// so the B fragment a lane needs (elements k0 and k0+1 at column n, with
// k0 = kk + 2*khalf) is ONE contiguous ds_load_b64; the per-lane base
// (khalf*64 + n)*8 is loop-invariant and every (kk,j) term folds into the
// 16-bit DS immediate offset. inv_deg (mean normalizer) is folded into the
// A-fragment load of the agg half. All control flow is wave-uniform, so
// EXEC is all-ones at every WMMA (ISA requirement).
// ---------------------------------------------------------------------------
__global__ void __launch_bounds__(256)
sage_linear_wmma_kernel(const float* __restrict__ h,
                        const float* __restrict__ agg,
                        const float* __restrict__ inv_deg,
                        const float* __restrict__ W,   // [128,64] row-major
                        const float* __restrict__ bias,
                        float* __restrict__ out,
                        int n, int apply_relu) {
    __shared__ float Wlds[DIM2 * DIM];   // 32 KB, holds K-pair-packed W

    // Stage + repack W: element (k, n) -> Wlds[(k>>1)*128 + n*2 + (k&1)]
    for (int i = threadIdx.x; i < DIM2 * DIM; i += 256) {
        int k = i >> 6;          // 0..127
        int c = i & 63;          // 0..63
        Wlds[(k >> 1) * 128 + c * 2 + (k & 1)] = W[i];
    }
    __syncthreads();

    const int lane  = threadIdx.x & 31;
    const int wave  = threadIdx.x >> 5;
    const int m16   = lane & 15;     // row within tile (A/C) or col (B/C)
    const int khalf = lane >> 4;     // selects K pair within A/B fragment

    const int rowBase = blockIdx.x * 128 + wave * 16;   // 16 rows per wave
    int row  = rowBase + m16;
    int rowc = row < n ? row : (n - 1);                 // clamp (EXEC stays full)

    const float idg = inv_deg[rowc];
    const float* __restrict__ hrow = h   + (size_t)rowc * DIM + 2 * khalf;
    const float* __restrict__ arow = agg + (size_t)rowc * DIM + 2 * khalf;

    // Per-lane base into packed W (v2f view): (kk/2 + khalf)*64 + j*16 + n
    const v2f* __restrict__ Wp = (const v2f*)Wlds + khalf * 64 + m16;

    // Accumulators init to bias (same value down each column of C)
    v8f acc[4];
#pragma unroll
    for (int j = 0; j < 4; ++j) {
        float bv = bias[j * 16 + m16];
        acc[j] = (v8f){bv, bv, bv, bv, bv, bv, bv, bv};
    }

    // ---- Self half: K = 0..63 ------------------------------------------
#pragma unroll
    for (int kk = 0; kk < DIM; kk += 4) {
        v2f a = *(const v2f*)(hrow + kk);              // global_load_b64
#pragma unroll
        for (int j = 0; j < 4; ++j) {
            v2f bf = Wp[(kk >> 1) * 64 + j * 16];      // ds_load_b64, imm offs
            acc[j] = __builtin_amdgcn_wmma_f32_16x16x4_f32(
                false, a, false, bf, (short)0, acc[j], false, false);
        }
    }

    // ---- Neighbor-mean half: K = 64..127 (fold inv_deg) ----------------
#pragma unroll
    for (int kk = 0; kk < DIM; kk += 4) {
        v2f a = *(const v2f*)(arow + kk);
        a.x *= idg;
        a.y *= idg;
#pragma unroll
        for (int j = 0; j < 4; ++j) {
            v2f bf = Wp[((kk + DIM) >> 1) * 64 + j * 16];
            acc[j] = __builtin_amdgcn_wmma_f32_16x16x4_f32(
                false, a, false, bf, (short)0, acc[j], false, false);
        }
    }

    // Optional ReLU
    if (apply_relu) {
#pragma unroll
        for (int j = 0; j < 4; ++j)
#pragma unroll
            for (int r = 0; r < 8; ++r)
                acc[j][r] = fmaxf(acc[j][r], 0.0f);
    }

    // Store C/D layout: m = rowBase + r + 8*khalf, col = j*16 + m16
#pragma unroll
    for (int r = 0; r < 8; ++r) {
        int m = rowBase + r + 8 * khalf;
        if (m < n) {
            float* o = out + (size_t)m * DIM + m16;
#pragma unroll
            for (int j = 0; j < 4; ++j)
                o[j * 16] = acc[j][r];
        }
    }
}

// ---------------------------------------------------------------------------
// Launcher
// ---------------------------------------------------------------------------
extern "C" void kernel_launch(void* const* d_in, const int* in_sizes, int n_in,
                              void* d_out, int out_size, void* d_ws, size_t ws_size,
                              hipStream_t stream) {
    const float* x    = (const float*)d_in[0];
    const int*   ei   = (const int*)d_in[1];
    const float* W0   = (const float*)d_in[2];
    const float* b0   = (const float*)d_in[3];
    const float* W1   = (const float*)d_in[4];
    const float* b1   = (const float*)d_in[5];

    const int n      = in_sizes[0] / DIM;      // 100000 nodes
    const int nedges = in_sizes[1] / 2;        // 2000000 edges
    const int* src = ei;                       // edge_index[0]
    const int* dst = ei + nedges;              // edge_index[1]

    // Workspace layout
    float* inv_deg = (float*)d_ws;                      // n floats
    float* agg     = inv_deg + n;                       // n*64 floats
    float* h1      = agg + (size_t)n * DIM;             // n*64 floats
    float* outp    = (float*)d_out;

    const int TB = 256;
    const int gridN    = (n + TB - 1) / TB;
    const int gridFill = 2048;
    const int gridScat = 4096;
    const int gridGemm = (n + 127) / 128;

    // Degree -> inv_deg (shared by both layers)
    fill_zero_kernel<<<gridN, TB, 0, stream>>>(inv_deg, (long)n);
    degree_kernel<<<2048, TB, 0, stream>>>(dst, inv_deg, nedges);
    invdeg_kernel<<<gridN, TB, 0, stream>>>(inv_deg, n);

    // ---- Layer 0: agg = scatter(x); h1 = relu(concat(x, agg/deg) @ W0 + b0)
    fill_zero_kernel<<<gridFill, TB, 0, stream>>>(agg, (long)n * DIM);
    scatter_kernel<<<gridScat, TB, 0, stream>>>(x, src, dst, agg, nedges);
    sage_linear_wmma_kernel<<<gridGemm, TB, 0, stream>>>(x, agg, inv_deg, W0, b0,
                                                         h1, n, 1);

    // ---- Layer 1: agg = scatter(h1); out = concat(h1, agg/deg) @ W1 + b1
    fill_zero_kernel<<<gridFill, TB, 0, stream>>>(agg, (long)n * DIM);
    scatter_kernel<<<gridScat, TB, 0, stream>>>(h1, src, dst, agg, nedges);
    sage_linear_wmma_kernel<<<gridGemm, TB, 0, stream>>>(h1, agg, inv_deg, W1, b1,
                                                         outp, n, 0);
}